// PSP_HDC_DUC_89206470738670
// MI455X (gfx1250) — compile-verified
//
#include <hip/hip_runtime.h>
#include <stdint.h>

#ifndef __has_builtin
#define __has_builtin(x) 0
#endif

#define NCLS 6
#define NCH (NCLS * 64)            // 384 input channels
#define OUTHW 304
#define PLANE (OUTHW * OUTHW)      // 92416
#define LDS_STRIDE 36              // 32 + 4 pad floats: kills LDS bank conflicts, keeps 16B align

typedef float f4 __attribute__((ext_vector_type(4)));
typedef int   v4i __attribute__((__vector_size__(4 * sizeof(int))));

#if defined(__gfx1250__) && __has_builtin(__builtin_amdgcn_global_load_async_to_lds_b128)
#define USE_ASYNC 1
#else
#define USE_ASYNC 0
#endif

// One block per (b, c, i): stage 64 channels x 32 floats of row i in LDS,
// emit 8 output rows x 256 cols as b128 stores.
__global__ __launch_bounds__(256) void duc_d2s(const float* __restrict__ x,
                                               float* __restrict__ out) {
    __shared__ float lds[64 * LDS_STRIDE];  // 9216 B
    const int t   = threadIdx.x;
    const int blk = blockIdx.x;
    const int i   = blk & 31;               // input row
    const int c   = (blk >> 5) % NCLS;      // class
    const int b   = blk / (32 * NCLS);      // batch

    // base of row i of channel c*64
    const float* src = x + (((size_t)b * NCH + (size_t)c * 64) * 32 + i) * 32;

    // ---- load phase: 512 chunks of 16B. chunk k: chan = k>>3, jstart = (k&7)*4
#if USE_ASYNC
    {
        __attribute__((address_space(3))) float* lds3 =
            (__attribute__((address_space(3))) float*)lds;
#pragma unroll
        for (int k = t; k < 512; k += 256) {
            const int chan = k >> 3;
            const int jst  = (k & 7) << 2;
            __builtin_amdgcn_global_load_async_to_lds_b128(
                (__attribute__((address_space(1))) v4i*)(uintptr_t)(src + (size_t)chan * 1024 + jst),
                (__attribute__((address_space(3))) v4i*)(lds3 + chan * LDS_STRIDE + jst),
                0, 0);
        }
#if __has_builtin(__builtin_amdgcn_s_wait_asynccnt)
        __builtin_amdgcn_s_wait_asynccnt(0);
#else
        asm volatile("s_wait_asynccnt 0" ::: "memory");
#endif
    }
#else
#pragma unroll
    for (int k = t; k < 512; k += 256) {
        const int chan = k >> 3;
        const int jst  = (k & 7) << 2;
        *(f4*)(lds + chan * LDS_STRIDE + jst) =
            *(const f4*)(src + (size_t)chan * 1024 + jst);
    }
#endif
    __syncthreads();

    // ---- store phase: 512 chunks of 16B across 8 output rows.
    // chunk m: p = m>>6 (sub-row), m0 = m&63 -> cols [4*m0, 4*m0+3]
    // col = j*8+q with j = m0>>1, q = (m0&1)*4 + e  =>  gather stride LDS_STRIDE
    float* dst_plane = out + (size_t)(b * NCLS + c) * PLANE;
#pragma unroll
    for (int m = t; m < 512; m += 256) {
        const int p    = m >> 6;
        const int m0   = m & 63;
        const int j    = m0 >> 1;
        const int qb   = (m0 & 1) << 2;
        const int base = (p * 8 + qb) * LDS_STRIDE + j;
        f4 v;
        v.x = lds[base];
        v.y = lds[base + LDS_STRIDE];
        v.z = lds[base + 2 * LDS_STRIDE];
        v.w = lds[base + 3 * LDS_STRIDE];
        const int row = i * 8 + p;
        *(f4*)(dst_plane + (size_t)row * OUTHW + (m0 << 2)) = v;
    }
}

// Zero only the border: per plane, rows 0..255 cols 256..303 (256*12 float4)
// plus rows 256..303 full width (48*76 float4) -> 6720 float4 per plane.
#define BORDER_PER_PLANE 6720
#define NPLANES (64 * NCLS)                           // 384
#define BORDER_TOTAL (NPLANES * BORDER_PER_PLANE)     // 2,580,480 float4 stores

__global__ __launch_bounds__(256) void duc_border(float* __restrict__ out) {
    const int idx = blockIdx.x * 256 + threadIdx.x;
    if (idx >= BORDER_TOTAL) return;
    const int plane = idx / BORDER_PER_PLANE;
    const int r     = idx - plane * BORDER_PER_PLANE;
    int row, col;
    if (r < 3072) {                     // right strip: rows 0..255, cols 256..303
        row = r / 12;
        col = 256 + (r - row * 12) * 4;
    } else {                            // bottom strip: rows 256..303, full width
        const int r2 = r - 3072;
        const int rr = r2 / 76;
        row = 256 + rr;
        col = (r2 - rr * 76) * 4;
    }
    const f4 z = {0.f, 0.f, 0.f, 0.f};
    *(f4*)(out + (size_t)plane * PLANE + (size_t)row * OUTHW + col) = z;
}

extern "C" void kernel_launch(void* const* d_in, const int* in_sizes, int n_in,
                              void* d_out, int out_size, void* d_ws, size_t ws_size,
                              hipStream_t stream) {
    const float* x = (const float*)d_in[0];
    float* out = (float*)d_out;

    // border: 2,580,480 threads / 256 = 10080 blocks (exact)
    duc_border<<<BORDER_TOTAL / 256, 256, 0, stream>>>(out);
    // main shuffle: 64 * 6 * 32 = 12288 blocks
    duc_d2s<<<64 * NCLS * 32, 256, 0, stream>>>(x, out);
}